// EKTA_74268574483055
// MI455X (gfx1250) — compile-verified
//
#include <hip/hip_runtime.h>
#include <math.h>

#define T_LEN   1024
#define KCNT    1024
#define H_DIM   256
#define E_DIM   64
#define TOPIC   100
#define EXLEN   768
#define GRU_IN  201     // 2*TOPIC + 1
#define TOPK    64
#define G3H     768     // 3*H

typedef __attribute__((ext_vector_type(2))) float v2f;
typedef __attribute__((ext_vector_type(4))) float v4f;
typedef __attribute__((ext_vector_type(8))) float v8f;

// workspace layout (float offsets)
#define WS_V     0                         // 100 (padded to 128)
#define WS_IDX   192                       // 64 ints
#define WS_BETA  256                       // 64
#define WS_ALPHA 320                       // 1024
#define WS_WX    1344                      // 768
#define WS_ATTN  2112                      // 262144 (K*H)
#define WS_GH    (2112 + KCNT * H_DIM)     // 786432 (K*3H)
// total ~1,050,688 floats (~4.2 MB)

// ---------------------------------------------------------------------------
// Kernel A: v, kn, beta_all, top-64 + softmax, alpha softmax, x, wx = W_ih@x
// ---------------------------------------------------------------------------
__global__ void ekta_setup(const float* __restrict__ co_e,
                           const float* __restrict__ ex_e,
                           const float* __restrict__ s_in,
                           const float* __restrict__ vs,
                           const float* __restrict__ W_resize,
                           const float* __restrict__ b_resize,
                           const float* __restrict__ Wk,
                           const float* __restrict__ bk,
                           const float* __restrict__ km,
                           const float* __restrict__ W_ih,
                           float* __restrict__ ws) {
    __shared__ float s_v[TOPIC];
    __shared__ float s_kn[E_DIM];
    __shared__ float s_buf[1024];
    __shared__ float s_x[GRU_IN];
    __shared__ float s_topv[TOPK];
    __shared__ int   s_topi[TOPK];
    __shared__ float cand_v[256];
    __shared__ int   cand_i[256];
    __shared__ float s_scalar;

    const int tid = threadIdx.x;

    // v = W_resize @ ex_e + b_resize   [100]
    if (tid < TOPIC) {
        float acc = b_resize[tid];
        const float* row = W_resize + tid * EXLEN;
        for (int i = 0; i < EXLEN; ++i) acc += row[i] * ex_e[i];
        s_v[tid] = acc;
        ws[WS_V + tid] = acc;
    }
    // kn = Wk @ co_e + bk              [64]
    if (tid < E_DIM) {
        float acc = bk[tid];
        const float* row = Wk + tid * KCNT;
        for (int i = 0; i < KCNT; ++i) acc += row[i] * co_e[i];
        s_kn[tid] = acc;
    }
    __syncthreads();

    // beta_all = vs @ v                [1024]
    for (int r = 0; r < 4; ++r) {
        int t = tid * 4 + r;
        float acc = 0.f;
        const float* row = vs + t * TOPIC;
        for (int i = 0; i < TOPIC; ++i) acc += row[i] * s_v[i];
        s_buf[t] = acc;
    }
    __syncthreads();

    // top-64 selection (iterative argmax: per-thread scan + serial combine)
    for (int it = 0; it < TOPK; ++it) {
        float bv = -3.0e38f; int bi = 0;
        for (int r = 0; r < 4; ++r) {
            int t = tid * 4 + r;
            float vv = s_buf[t];
            if (vv > bv) { bv = vv; bi = t; }
        }
        cand_v[tid] = bv; cand_i[tid] = bi;
        __syncthreads();
        if (tid == 0) {
            float mv = cand_v[0]; int mi = cand_i[0];
            for (int t = 1; t < 256; ++t)
                if (cand_v[t] > mv) { mv = cand_v[t]; mi = cand_i[t]; }
            s_topv[it] = mv; s_topi[it] = mi;
            s_buf[mi] = -3.0e38f;
        }
        __syncthreads();
    }

    // beta = softmax(top vals); emit beta + idx (thread 0, 64 elems)
    if (tid == 0) {
        float m = s_topv[0];
        for (int j = 1; j < TOPK; ++j) m = fmaxf(m, s_topv[j]);
        float sum = 0.f;
        for (int j = 0; j < TOPK; ++j) { float e = expf(s_topv[j] - m); s_topv[j] = e; sum += e; }
        float inv = 1.f / sum;
        for (int j = 0; j < TOPK; ++j) {
            ws[WS_BETA + j] = s_topv[j] * inv;
            ((int*)ws)[WS_IDX + j] = s_topi[j];
        }
    }

    // alpha = softmax(km @ kn)         [1024]  (reuse s_buf)
    for (int r = 0; r < 4; ++r) {
        int t = tid * 4 + r;
        float acc = 0.f;
        const float* row = km + t * E_DIM;
        for (int i = 0; i < E_DIM; ++i) acc += row[i] * s_kn[i];
        s_buf[t] = acc;
    }
    __syncthreads();
    if (tid == 0) {
        float m = s_buf[0];
        for (int t = 1; t < 1024; ++t) m = fmaxf(m, s_buf[t]);
        s_scalar = m;
    }
    __syncthreads();
    {
        float m = s_scalar;
        float part = 0.f;
        for (int r = 0; r < 4; ++r) {
            int t = tid * 4 + r;
            float e = expf(s_buf[t] - m);
            s_buf[t] = e;
            part += e;
        }
        cand_v[tid] = part;
    }
    __syncthreads();
    if (tid == 0) {
        float sum = 0.f;
        for (int t = 0; t < 256; ++t) sum += cand_v[t];
        s_scalar = 1.f / sum;
    }
    __syncthreads();
    for (int r = 0; r < 4; ++r) {
        int t = tid * 4 + r;
        ws[WS_ALPHA + t] = s_buf[t] * s_scalar;
    }

    // x = [v*mask, v*(1-mask), s]      [201]
    float sc = s_in[0];
    float mask = (sc >= 0.5f) ? 1.f : 0.f;
    if (tid < TOPIC) {
        s_x[tid]         = s_v[tid] * mask;
        s_x[TOPIC + tid] = s_v[tid] * (1.f - mask);
    }
    if (tid == 0) s_x[2 * TOPIC] = sc;
    __syncthreads();

    // wx = W_ih @ x                    [768]  (rank-1 trick: gi = alpha*wx + b_ih)
    for (int r = 0; r < 3; ++r) {
        int j = tid + r * 256;
        float acc = 0.f;
        const float* row = W_ih + j * GRU_IN;
        for (int i = 0; i < GRU_IN; ++i) acc += row[i] * s_x[i];
        ws[WS_WX + j] = acc;
    }
}

// ---------------------------------------------------------------------------
// Kernel B: attn_h = sum_j beta[j] * hs[idx[j], :, :]   -- 67 MB HBM, the
// roofline-dominant kernel. float4 (b128) coalesced loads.
// ---------------------------------------------------------------------------
__global__ void ekta_attn(const float* __restrict__ hs, float* __restrict__ ws) {
    __shared__ float s_beta[TOPK];
    __shared__ int   s_idx[TOPK];
    const int tid = threadIdx.x;
    if (tid < TOPK) {
        s_beta[tid] = ws[WS_BETA + tid];
        s_idx[tid]  = ((const int*)ws)[WS_IDX + tid];
    }
    __syncthreads();
    const int gid = blockIdx.x * blockDim.x + tid;   // float4 index, 65536 total
    v4f acc = {0.f, 0.f, 0.f, 0.f};
    for (int j = 0; j < TOPK; ++j) {
        const v4f* p = (const v4f*)(hs + (size_t)s_idx[j] * (KCNT * H_DIM));
        v4f d = p[gid];
        acc += d * s_beta[j];
    }
    ((v4f*)(ws + WS_ATTN))[gid] = acc;
}

// ---------------------------------------------------------------------------
// Kernel C: hkp = alpha @ attn_h.reshape(K,H); score -> out[0]
// ---------------------------------------------------------------------------
__global__ void ekta_hkp_score(const float* __restrict__ W_score,
                               const float* __restrict__ b_score,
                               const float* __restrict__ ws,
                               float* __restrict__ out) {
    __shared__ float s_alpha[KCNT];
    __shared__ float s_part[256];
    const int tid = threadIdx.x;
    for (int r = 0; r < 4; ++r) s_alpha[tid * 4 + r] = ws[WS_ALPHA + tid * 4 + r];
    __syncthreads();
    const float* attn = ws + WS_ATTN;
    float acc = 0.f;
    for (int kk = 0; kk < KCNT; ++kk) acc += s_alpha[kk] * attn[kk * H_DIM + tid];
    // score partial: W_score layout = [v part (100) | hkp part (256)]
    float p = W_score[TOPIC + tid] * acc;
    if (tid < TOPIC) p += W_score[tid] * ws[WS_V + tid];
    s_part[tid] = p;
    __syncthreads();
    if (tid == 0) {
        float sum = b_score[0];
        for (int t = 0; t < 256; ++t) sum += s_part[t];
        out[0] = sum;
    }
}

// ---------------------------------------------------------------------------
// Kernel D: gh = h0 @ W_hh.T  [1024 x 768] via V_WMMA_F32_16X16X4_F32.
// One wave per 16x16 tile; 64 wmma steps over the K=256 reduction.
// A 16x4 layout: lane L holds row M=L%16, K = (L/16)*2 + {0,1}  (b64 load)
// B 4x16  layout: lane L holds col N=L%16, K = (L/16)*2 + {0,1}  (b64 load)
// C/D: vgpr r -> (M = r + (L/16)*8, N = L%16)
// ---------------------------------------------------------------------------
__global__ void ekta_gemm_wmma(const float* __restrict__ h0,
                               const float* __restrict__ W_hh,
                               float* __restrict__ gh) {
    const int wave = (blockIdx.x * blockDim.x + threadIdx.x) >> 5;  // 0..3071
    const int lane = threadIdx.x & 31;
    const int kt = wave / 48;          // row tile 0..63
    const int jt = wave % 48;          // col tile 0..47
    const int m0 = kt * 16;
    const int n0 = jt * 16;
    const int half = lane >> 4;
    const int lm = lane & 15;

    const float* Arow = h0   + (size_t)(m0 + lm) * H_DIM + half * 2;
    const float* Brow = W_hh + (size_t)(n0 + lm) * H_DIM + half * 2;

    v8f acc = {};
#pragma unroll 8
    for (int c0 = 0; c0 < H_DIM; c0 += 4) {
        v2f a = *(const v2f*)(Arow + c0);
        v2f b = *(const v2f*)(Brow + c0);
        acc = __builtin_amdgcn_wmma_f32_16x16x4_f32(
            /*neg_a=*/false, a, /*neg_b=*/false, b,
            /*c_mod=*/(short)0, acc, /*reuse_a=*/false, /*reuse_b=*/false);
    }

    float* orow = gh + (size_t)(m0 + half * 8) * G3H + n0 + lm;
#pragma unroll
    for (int r = 0; r < 8; ++r) orow[(size_t)r * G3H] = acc[r];
}

// ---------------------------------------------------------------------------
// Kernel E: GRU gates + h_new -> out[1:]
// gi[k,j] = alpha[k]*wx[j] + b_ih[j] ; gh already holds h0 @ W_hh.T
// ---------------------------------------------------------------------------
__global__ void ekta_gru_gates(const float* __restrict__ h0,
                               const float* __restrict__ b_ih,
                               const float* __restrict__ b_hh,
                               const float* __restrict__ ws,
                               float* __restrict__ out) {
    const int k = blockIdx.x;
    const int c = threadIdx.x;
    const float a = ws[WS_ALPHA + k];
    const float* wx = ws + WS_WX;
    const float* gh = ws + WS_GH + (size_t)k * G3H;

    float ir = a * wx[c]       + b_ih[c];
    float iz = a * wx[256 + c] + b_ih[256 + c];
    float in_ = a * wx[512 + c] + b_ih[512 + c];
    float hr = gh[c]       + b_hh[c];
    float hz = gh[256 + c] + b_hh[256 + c];
    float hn = gh[512 + c] + b_hh[512 + c];

    float r = 1.f / (1.f + expf(-(ir + hr)));
    float z = 1.f / (1.f + expf(-(iz + hz)));
    float n = tanhf(in_ + r * hn);
    float h0v = h0[(size_t)k * H_DIM + c];
    out[1 + (size_t)k * H_DIM + c] = (1.f - z) * n + z * h0v;
}

// ---------------------------------------------------------------------------
extern "C" void kernel_launch(void* const* d_in, const int* in_sizes, int n_in,
                              void* d_out, int out_size, void* d_ws, size_t ws_size,
                              hipStream_t stream) {
    (void)in_sizes; (void)n_in; (void)out_size; (void)ws_size;
    const float* co_e     = (const float*)d_in[0];
    const float* ex_e     = (const float*)d_in[1];
    const float* s        = (const float*)d_in[2];
    const float* h        = (const float*)d_in[3];
    const float* vs       = (const float*)d_in[4];
    const float* hs       = (const float*)d_in[5];
    const float* W_resize = (const float*)d_in[6];
    const float* b_resize = (const float*)d_in[7];
    const float* Wk       = (const float*)d_in[8];
    const float* bk       = (const float*)d_in[9];
    const float* km       = (const float*)d_in[10];
    const float* W_score  = (const float*)d_in[11];
    const float* b_score  = (const float*)d_in[12];
    const float* W_ih     = (const float*)d_in[13];
    const float* W_hh     = (const float*)d_in[14];
    const float* b_ih     = (const float*)d_in[15];
    const float* b_hh     = (const float*)d_in[16];
    float* out = (float*)d_out;
    float* ws  = (float*)d_ws;

    ekta_setup<<<1, 256, 0, stream>>>(co_e, ex_e, s, vs, W_resize, b_resize,
                                      Wk, bk, km, W_ih, ws);
    ekta_attn<<<256, 256, 0, stream>>>(hs, ws);
    ekta_hkp_score<<<1, 256, 0, stream>>>(W_score, b_score, ws, out);
    ekta_gemm_wmma<<<384, 256, 0, stream>>>(h, W_hh, ws + WS_GH);
    ekta_gru_gates<<<KCNT, 256, 0, stream>>>(h, b_ih, b_hh, ws, out);
}